// GeATLayerWithSingleHead_43061342110001
// MI455X (gfx1250) — compile-verified
//
#include <hip/hip_runtime.h>
#include <hip/hip_bf16.h>
#include <math.h>

// ---- problem constants (match reference) ----
#define BB 32
#define NN 256
#define DD 128
#define TT 8
#define NEG_SLOPE 0.2f

typedef __attribute__((ext_vector_type(2))) float v2f;
typedef __attribute__((ext_vector_type(8))) float v8f;

// D = A(16x4) * B(4x16) + C, full fp32 (CDNA5 V_WMMA_F32_16X16X4_F32)
__device__ __forceinline__ v8f wmma_f32_4(v2f a, v2f b, v8f c) {
  return __builtin_amdgcn_wmma_f32_16x16x4_f32(
      /*neg_a=*/false, a, /*neg_b=*/false, b,
      /*c_mod=*/(short)0, c, /*reuse_a=*/false, /*reuse_b=*/false);
}

// =====================================================================
// Kernel 0: aT[t][e][k] = a[k][e][t]  -- makes score-stage B-fragments
// K-contiguous (single b64 load per WMMA instead of 2 scattered b32).
// grid.x = T*D, block = 128.
// =====================================================================
__global__ __launch_bounds__(128)
void geat_transposeA_kernel(const float* __restrict__ A, float* __restrict__ aT) {
  const int t = blockIdx.x / DD;
  const int e = blockIdx.x % DD;
  const int k = threadIdx.x;
  aT[((size_t)t * DD + e) * DD + k] = A[((size_t)k * DD + e) * TT + t];
}

// =====================================================================
// Kernel 1: src/dst/val projections.  out = x @ W^T + bias.
// val is stored TRANSPOSED (B,D,N) so kernel 4's B-frags are K-contiguous.
// grid.x = 3 * B * (N/16), block = 256 (8 waves; wave w -> 16-col tile w)
// =====================================================================
__global__ __launch_bounds__(256)
void geat_proj_kernel(const float* __restrict__ x,
                      const float* __restrict__ qw, const float* __restrict__ qb,
                      const float* __restrict__ kw, const float* __restrict__ kb,
                      const float* __restrict__ vw, const float* __restrict__ vb,
                      float* __restrict__ srcO, float* __restrict__ dstO,
                      float* __restrict__ valT) {
  const int blk   = blockIdx.x;
  const int which = blk % 3;
  const int tile  = blk / 3;
  const int b     = tile / (NN / 16);
  const int n0    = (tile % (NN / 16)) * 16;

  const float* W; const float* bias; float* out;
  if (which == 0)      { W = qw; bias = qb; out = srcO; }
  else if (which == 1) { W = kw; bias = kb; out = dstO; }
  else                 { W = vw; bias = vb; out = valT; }

  const int lane = threadIdx.x & 31;
  const int wave = threadIdx.x >> 5;
  const int hi   = lane >> 4;    // which K-half / which M-half
  const int lo   = lane & 15;
  const int col  = wave * 16 + lo;

  const float* xt = x + ((size_t)b * NN + n0) * DD;

  v8f acc = {};
  for (int k = 0; k < DD; k += 4) {
    const int kk = k + 2 * hi;
    v2f a, bb_;
    a.x  = xt[lo * DD + kk];
    a.y  = xt[lo * DD + kk + 1];
    bb_.x = W[col * DD + kk];      // (W^T)[k][col] = W[col][k]
    bb_.y = W[col * DD + kk + 1];
    acc = wmma_f32_4(a, bb_, acc);
  }

  const float bv = bias[col];
  if (which == 2) {                // val: store transposed (B, D, N)
#pragma unroll
    for (int i = 0; i < 8; ++i) {
      const int row = n0 + i + 8 * hi;
      out[((size_t)b * DD + col) * NN + row] = acc[i] + bv;
    }
  } else {                         // src/dst: row-major (B, N, D)
#pragma unroll
    for (int i = 0; i < 8; ++i) {
      const int row = n0 + i + 8 * hi;
      out[((size_t)b * NN + row) * DD + col] = acc[i] + bv;
    }
  }
}

// =====================================================================
// Kernel 2: scores + edge-type gather + mask + leaky ReLU.
//   srcA_t = src_tile(16xD) @ a[:,:,t]      (wave t, into LDS)
//   scores_t(16x16) = srcA_t @ dst_tile^T   (8 WMMAs share one B-frag)
//   s[b,n,m] = leakyrelu(mask(scores_{edge[b,n,m]}[n,m]))
// grid.x = B * (N/16), block = 256
// =====================================================================
__global__ __launch_bounds__(256)
void geat_scores_kernel(const float* __restrict__ srcI,
                        const float* __restrict__ dstI,
                        const float* __restrict__ aT,     // (T,E,K)
                        const int*   __restrict__ edges,  // (B,N,N)
                        float* __restrict__ sOut) {       // (B,N,N)
  __shared__ float lds_srcA[TT][16][DD];   // 64 KB

  const int b    = blockIdx.x / (NN / 16);
  const int n0   = (blockIdx.x % (NN / 16)) * 16;
  const int lane = threadIdx.x & 31;
  const int wave = threadIdx.x >> 5;
  const int hi   = lane >> 4;
  const int lo   = lane & 15;

  // ---- phase 1: wave t computes srcA_t (16 x 128) ----
  {
    const int t = wave;
    const float* st = srcI + ((size_t)b * NN + n0) * DD;
    const float* At = aT + (size_t)t * DD * DD;   // At[e][k] = a[k][e][t]
    v8f acc[8];
    v8f zero = {};
#pragma unroll
    for (int et = 0; et < 8; ++et) acc[et] = zero;

    for (int k = 0; k < DD; k += 4) {
      const int kk = k + 2 * hi;
      v2f a;
      a.x = st[lo * DD + kk];
      a.y = st[lo * DD + kk + 1];
#pragma unroll
      for (int et = 0; et < 8; ++et) {
        const int e = et * 16 + lo;
        v2f bb_;                        // B[kk][e] = At[e][kk] (contiguous pair)
        bb_.x = At[(size_t)e * DD + kk];
        bb_.y = At[(size_t)e * DD + kk + 1];
        acc[et] = wmma_f32_4(a, bb_, acc[et]);
      }
    }
#pragma unroll
    for (int et = 0; et < 8; ++et)
#pragma unroll
      for (int i = 0; i < 8; ++i)
        lds_srcA[t][i + 8 * hi][et * 16 + lo] = acc[et][i];
  }
  __syncthreads();

  // ---- phase 2: each wave covers m-tiles wave, wave+8 ----
  for (int mt = wave; mt < NN / 16; mt += 8) {
    const int m = mt * 16 + lo;
    v8f sc[TT];
    v8f zero = {};
#pragma unroll
    for (int t = 0; t < TT; ++t) sc[t] = zero;

    for (int k = 0; k < DD; k += 4) {   // contract over e
      const int kk = k + 2 * hi;
      v2f bb_;                          // dst^T[e][m] = dst[m][e]
      bb_.x = dstI[((size_t)b * NN + m) * DD + kk];
      bb_.y = dstI[((size_t)b * NN + m) * DD + kk + 1];
#pragma unroll
      for (int t = 0; t < TT; ++t) {
        v2f a;
        a.x = lds_srcA[t][lo][kk];
        a.y = lds_srcA[t][lo][kk + 1];
        sc[t] = wmma_f32_4(a, bb_, sc[t]);
      }
    }

#pragma unroll
    for (int i = 0; i < 8; ++i) {
      const int rn = n0 + i + 8 * hi;
      const int e  = edges[((size_t)b * NN + rn) * NN + m];
      const int tt = e < 0 ? 0 : e;     // jnp.clip(edges, 0)
      float sv = 0.f;
#pragma unroll
      for (int t = 0; t < TT; ++t)
        if (tt == t) sv = sc[t][i];
      if (e == -1) sv = -__builtin_inff();
      sv = sv > 0.f ? sv : NEG_SLOPE * sv;      // leaky relu (-inf stays -inf)
      sOut[((size_t)b * NN + rn) * NN + m] = sv;
    }
  }
}

// =====================================================================
// Kernel 3: row softmax over m (in place). grid.x = B*N, block = 256.
// wave32 shfl reductions + 8-slot LDS combine.
// =====================================================================
__global__ __launch_bounds__(256)
void geat_softmax_kernel(float* __restrict__ s) {
  __shared__ float rmax[8];
  __shared__ float rsum[8];
  const int tid  = threadIdx.x;
  const int lane = tid & 31;
  const int wave = tid >> 5;
  float* row = s + (size_t)blockIdx.x * NN;

  float v = row[tid];
  float m = v;
#pragma unroll
  for (int off = 16; off > 0; off >>= 1)
    m = fmaxf(m, __shfl_xor(m, off, 32));
  if (lane == 0) rmax[wave] = m;
  __syncthreads();
  float mx = rmax[0];
#pragma unroll
  for (int i = 1; i < 8; ++i) mx = fmaxf(mx, rmax[i]);

  float e = expf(v - mx);    // exp(-inf - mx) = 0
  float sm = e;
#pragma unroll
  for (int off = 16; off > 0; off >>= 1)
    sm += __shfl_xor(sm, off, 32);
  if (lane == 0) rsum[wave] = sm;
  __syncthreads();
  float tot = 0.f;
#pragma unroll
  for (int i = 0; i < 8; ++i) tot += rsum[i];

  row[tid] = e / tot;
}

// =====================================================================
// Kernel 4: out = att @ val (16x256 @ 256x128) -> LDS -> @ pw^T + pb.
// val is consumed from its transposed (B,D,N) layout: K-contiguous b64.
// grid.x = B * (N/16), block = 256.
// =====================================================================
__global__ __launch_bounds__(256)
void geat_out_kernel(const float* __restrict__ att,    // (B,N,N)
                     const float* __restrict__ valT,   // (B,D,N)
                     const float* __restrict__ pw, const float* __restrict__ pb,
                     float* __restrict__ outF) {
  __shared__ float lds_out[16][DD];   // 8 KB

  const int b    = blockIdx.x / (NN / 16);
  const int n0   = (blockIdx.x % (NN / 16)) * 16;
  const int lane = threadIdx.x & 31;
  const int wave = threadIdx.x >> 5;
  const int hi   = lane >> 4;
  const int lo   = lane & 15;
  const int col  = wave * 16 + lo;

  // phase 1: att_tile(16x256) @ val(256x128); B[k][col] = valT[b][col][k]
  {
    const float* at = att + ((size_t)b * NN + n0) * NN;
    const float* vt = valT + ((size_t)b * DD + col) * NN;
    v8f acc = {};
    for (int k = 0; k < NN; k += 4) {
      const int kk = k + 2 * hi;
      v2f a, bb_;
      a.x  = at[lo * NN + kk];
      a.y  = at[lo * NN + kk + 1];
      bb_.x = vt[kk];
      bb_.y = vt[kk + 1];
      acc = wmma_f32_4(a, bb_, acc);
    }
#pragma unroll
    for (int i = 0; i < 8; ++i)
      lds_out[i + 8 * hi][col] = acc[i];
  }
  __syncthreads();

  // phase 2: (16x128) @ pw^T + pb
  {
    v8f acc = {};
    for (int k = 0; k < DD; k += 4) {
      const int kk = k + 2 * hi;
      v2f a, bb_;
      a.x  = lds_out[lo][kk];
      a.y  = lds_out[lo][kk + 1];
      bb_.x = pw[col * DD + kk];
      bb_.y = pw[col * DD + kk + 1];
      acc = wmma_f32_4(a, bb_, acc);
    }
    const float bv = pb[col];
#pragma unroll
    for (int i = 0; i < 8; ++i) {
      const int row = n0 + i + 8 * hi;
      outF[((size_t)b * NN + row) * DD + col] = acc[i] + bv;
    }
  }
}

// =====================================================================
extern "C" void kernel_launch(void* const* d_in, const int* in_sizes, int n_in,
                              void* d_out, int out_size, void* d_ws, size_t ws_size,
                              hipStream_t stream) {
  const float* x     = (const float*)d_in[0];   // (B,N,D)
  const int*   edges = (const int*)  d_in[1];   // (B,N,N)
  const float* A     = (const float*)d_in[2];   // (D,D,T)
  const float* qw    = (const float*)d_in[3];
  const float* qb    = (const float*)d_in[4];
  const float* kw    = (const float*)d_in[5];
  const float* kb    = (const float*)d_in[6];
  const float* vw    = (const float*)d_in[7];
  const float* vb    = (const float*)d_in[8];
  const float* pw    = (const float*)d_in[9];
  const float* pb    = (const float*)d_in[10];
  float* outF = (float*)d_out;

  // workspace layout (floats): src | dst | valT | s | aT   (~20.5 MB)
  const size_t BND = (size_t)BB * NN * DD;   // 1,048,576
  const size_t BNN = (size_t)BB * NN * NN;   // 2,097,152
  float* ws   = (float*)d_ws;
  float* srcW = ws;
  float* dstW = ws + BND;
  float* valW = ws + 2 * BND;                 // transposed (B,D,N)
  float* sW   = ws + 3 * BND;
  float* aTW  = ws + 3 * BND + BNN;           // (T,E,K) = 131,072 floats
  (void)ws_size; (void)in_sizes; (void)n_in; (void)out_size;

  geat_transposeA_kernel<<<dim3(TT * DD), dim3(128), 0, stream>>>(A, aTW);

  geat_proj_kernel<<<dim3(3 * BB * (NN / 16)), dim3(256), 0, stream>>>(
      x, qw, qb, kw, kb, vw, vb, srcW, dstW, valW);

  geat_scores_kernel<<<dim3(BB * (NN / 16)), dim3(256), 0, stream>>>(
      srcW, dstW, aTW, edges, sW);

  geat_softmax_kernel<<<dim3(BB * NN), dim3(256), 0, stream>>>(sW);

  geat_out_kernel<<<dim3(BB * (NN / 16)), dim3(256), 0, stream>>>(
      sW, valW, pw, pb, outF);
}